// PolyLoss_61942018343090
// MI455X (gfx1250) — compile-verified
//
#include <hip/hip_runtime.h>
#include <stdint.h>

#define NROWS 4096
#define V 128

typedef int v4i __attribute__((ext_vector_type(4)));
typedef __attribute__((address_space(3))) v4i lds_v4i;

// Async global -> LDS copy of 16 bytes per lane (gfx1250 async path, tracked
// on ASYNCcnt). Builtin takes int4-vector pointers per the clang diagnostic.
__device__ __forceinline__ void async_copy_b128(const float* __restrict__ g, float* l) {
#if __has_builtin(__builtin_amdgcn_global_load_async_to_lds_b128)
    __builtin_amdgcn_global_load_async_to_lds_b128(
        (v4i*)g, (lds_v4i*)l, /*offset=*/0, /*cpol=*/0);
#else
    uint32_t lds_off = (uint32_t)(uintptr_t)(lds_v4i*)l;
    asm volatile("global_load_async_to_lds_b128 %0, %1, off"
                 :: "v"(lds_off), "v"(g) : "memory");
#endif
}

__device__ __forceinline__ void wait_asynccnt0() {
#if __has_builtin(__builtin_amdgcn_s_wait_asynccnt)
    __builtin_amdgcn_s_wait_asynccnt(0);
#else
    asm volatile("s_wait_asynccnt 0" ::: "memory");
#endif
}

// One block (128 threads, 4 waves) per row n.
// Thread t computes the L1 sum for cyclic shift s = t, block min-reduces,
// writes min_s sum to row_min[n].
__global__ __launch_bounds__(128) void poly_row_min_kernel(
    const float* __restrict__ pred,   // [NROWS][V][2]
    const float* __restrict__ gt,     // [NROWS][V][2]
    float* __restrict__ row_min)      // [NROWS]
{
    __shared__ float2 sp[V];          // pred row
    __shared__ float2 sg[V];          // gt row
    __shared__ float  red[V];         // min-reduction scratch

    const int n = blockIdx.x;
    const int t = threadIdx.x;

    const float* prow = pred + (size_t)n * (V * 2);
    const float* grow = gt   + (size_t)n * (V * 2);

    // Stage both rows (2 x 1024 B) into LDS: 128 lanes x 16 B.
    // Waves 0-1 (lanes 0..63) fetch pred, waves 2-3 fetch gt.
    {
        const float* src;
        float*       dst;
        if (t < 64) { src = prow + t * 4;        dst = (float*)sp + t * 4; }
        else        { src = grow + (t - 64) * 4; dst = (float*)sg + (t - 64) * 4; }
        async_copy_b128(src, dst);
        wait_asynccnt0();      // each wave drains its own ASYNCcnt...
    }
    __syncthreads();           // ...then the barrier makes all LDS writes visible

    // Shift s = t: per_shift_sum = sum_j |p[(s+j)%V] - g[j]|_1
    float acc0 = 0.0f, acc1 = 0.0f;
#pragma unroll 8
    for (int j = 0; j < V; ++j) {
        const float2 g = sg[j];                 // uniform address -> LDS broadcast
        const float2 p = sp[(t + j) & (V - 1)]; // contiguous per wave -> conflict-free
        acc0 += fabsf(p.x - g.x);               // v_sub + v_add with |.| modifier
        acc1 += fabsf(p.y - g.y);
    }
    red[t] = acc0 + acc1;
    __syncthreads();

    // Min-tree over 128 shifts (deterministic).
    for (int w = V / 2; w > 0; w >>= 1) {
        if (t < w) red[t] = fminf(red[t], red[t + w]);
        __syncthreads();
    }
    if (t == 0) row_min[n] = red[0];
}

// Deterministic fixed-order sum of the 4096 per-row minima; applies the
// combined 1/V (mean over j) and 1/N (mean over rows) scaling.
__global__ __launch_bounds__(256) void poly_reduce_kernel(
    const float* __restrict__ row_min, float* __restrict__ out)
{
    __shared__ float s[256];
    const int t = threadIdx.x;
    float acc = 0.0f;
#pragma unroll
    for (int i = 0; i < NROWS / 256; ++i)       // 16 values per thread, fixed order
        acc += row_min[t + i * 256];
    s[t] = acc;
    __syncthreads();
    for (int w = 128; w > 0; w >>= 1) {
        if (t < w) s[t] += s[t + w];
        __syncthreads();
    }
    if (t == 0) out[0] = s[0] * (1.0f / ((float)V * (float)NROWS));
}

extern "C" void kernel_launch(void* const* d_in, const int* in_sizes, int n_in,
                              void* d_out, int out_size, void* d_ws, size_t ws_size,
                              hipStream_t stream) {
    const float* pred = (const float*)d_in[0];
    const float* gt   = (const float*)d_in[1];
    float* out = (float*)d_out;
    float* ws  = (float*)d_ws;      // needs NROWS * 4 = 16 KB

    poly_row_min_kernel<<<NROWS, V, 0, stream>>>(pred, gt, ws);
    poly_reduce_kernel<<<1, 256, 0, stream>>>(ws, out);
}